// FlexGCN_49881750176389
// MI455X (gfx1250) — compile-verified
//
#include <hip/hip_runtime.h>
#include <hip/hip_bf16.h>

typedef __attribute__((ext_vector_type(2))) float v2f;
typedef __attribute__((ext_vector_type(8))) float v8f;

#define Bn   32
#define Nn   307
#define Cin  64
#define Cout 64
#define Dd   10
#define Kk   4
#define ROWS (Bn * Nn)          // 9824 = 614 * 16
#define MT5  (ROWS / 16)        // 614 M-tiles, exact
#define AS_STRIDE 310           // even (8B rows) and 54r mod 64 injective for r<16
#define XG_STRIDE 194           // even (8B rows) and 2r mod 64 injective for r<16

// ---------------------------------------------------------------------------
// K1: attention2d  (avgpool over N -> fc1 -> relu -> fc2 -> softmax) -> att[B,4]
// ---------------------------------------------------------------------------
__global__ void attn_kernel(const float* __restrict__ x,
                            const float* __restrict__ fc1w, const float* __restrict__ fc1b,
                            const float* __restrict__ fc2w, const float* __restrict__ fc2b,
                            float* __restrict__ att) {
    __shared__ float pooled[Cin];
    __shared__ float h[Kk];
    __shared__ float logit[Kk];
    const int b = blockIdx.x;
    const int c = threadIdx.x;      // 0..63
    float s = 0.f;
    for (int n = 0; n < Nn; ++n) s += x[(b * Nn + n) * Cin + c];
    pooled[c] = s * (1.0f / (float)Nn);
    __syncthreads();
    if (c < Kk) {
        float acc = fc1b[c];
        for (int i = 0; i < Cin; ++i) acc += pooled[i] * fc1w[c * Cin + i];
        h[c] = fmaxf(acc, 0.f);
    }
    __syncthreads();
    if (c < Kk) {
        float acc = fc2b[c];
        for (int j = 0; j < Kk; ++j) acc += h[j] * fc2w[c * Kk + j];
        logit[c] = acc;
    }
    __syncthreads();
    if (c == 0) {
        float m = fmaxf(fmaxf(logit[0], logit[1]), fmaxf(logit[2], logit[3]));
        float e0 = __expf(logit[0] - m), e1 = __expf(logit[1] - m);
        float e2 = __expf(logit[2] - m), e3 = __expf(logit[3] - m);
        float inv = 1.0f / (e0 + e1 + e2 + e3);
        att[b * 4 + 0] = e0 * inv;
        att[b * 4 + 1] = e1 * inv;
        att[b * 4 + 2] = e2 * inv;
        att[b * 4 + 3] = e3 * inv;
    }
}

// ---------------------------------------------------------------------------
// K2: emb[b,n,d] = sum_k att[b,k] * node_embeddings[k,n,d]
// ---------------------------------------------------------------------------
__global__ void emb_kernel(const float* __restrict__ att, const float* __restrict__ ne,
                           float* __restrict__ emb) {
    const int idx = blockIdx.x * 256 + threadIdx.x;
    if (idx >= Bn * Nn * Dd) return;
    const int b  = idx / (Nn * Dd);
    const int nd = idx % (Nn * Dd);
    const float* a = att + b * 4;
    emb[idx] = a[0] * ne[nd]
             + a[1] * ne[1 * Nn * Dd + nd]
             + a[2] * ne[2 * Nn * Dd + nd]
             + a[3] * ne[3 * Nn * Dd + nd];
}

// ---------------------------------------------------------------------------
// K3: T2 = 2 * adj @ adj - I   (tiny: 5.8e7 FLOP, naive is fine)
// ---------------------------------------------------------------------------
__global__ void t2_kernel(const float* __restrict__ adj, float* __restrict__ T2) {
    const int idx = blockIdx.x * 256 + threadIdx.x;
    if (idx >= Nn * Nn) return;
    const int n = idx / Nn, m = idx % Nn;
    float acc = 0.f;
    for (int l = 0; l < Nn; ++l) acc += adj[n * Nn + l] * adj[l * Nn + m];
    T2[idx] = 2.0f * acc - (n == m ? 1.0f : 0.0f);
}

// ---------------------------------------------------------------------------
// K4a: x_g[.., k=0, :] = x  (identity support is a copy)
// ---------------------------------------------------------------------------
__global__ void xg0_kernel(const float* __restrict__ x, float* __restrict__ xg) {
    const int idx = blockIdx.x * 256 + threadIdx.x;
    if (idx >= ROWS * Cin) return;
    const int row = idx >> 6, c = idx & 63;
    xg[row * (3 * Cin) + c] = x[idx];
}

// ---------------------------------------------------------------------------
// K4b: WMMA supports-apply: xg[b,n,ks+1,c] = sum_m S[n,m] * x[b,m,c]
// grid: (20 mtiles, 2 supports, 32 batches); block 128 = 4 wave32.
// A-panel in LDS, zero-padded at K=307 so the hot loop has NO conditionals
// (no EXEC save/restore around the ds loads); K tail peeled with branchless
// address clamps on the B side only.
// ---------------------------------------------------------------------------
__global__ void spmm_wmma_kernel(const float* __restrict__ adj, const float* __restrict__ T2,
                                 const float* __restrict__ x, float* __restrict__ xg) {
    __shared__ __align__(16) float As[16][AS_STRIDE];  // rows 8B-aligned -> ds_load_b64
    const int mtile = blockIdx.x, ks = blockIdx.y, b = blockIdx.z;
    const float* __restrict__ S = (ks == 0) ? adj : T2;
    // stage 16 rows x cols 0..307 (col 307 = 0); clamp tail rows (masked at store)
    for (int idx = threadIdx.x; idx < 16 * 308; idx += 128) {
        int r = idx / 308, col = idx % 308;
        int row_g = mtile * 16 + r;
        row_g = row_g > (Nn - 1) ? (Nn - 1) : row_g;
        As[r][col] = (col < Nn) ? S[row_g * Nn + col] : 0.0f;
    }
    __syncthreads();

    const int tid = threadIdx.x;
    const int lane = tid & 31, wv = tid >> 5;
    const int half = lane >> 4;          // 0: K=k0,k0+1   1: K=k0+2,k0+3
    const int r = lane & 15;             // A row / B col within tile
    const int colbase = wv * 16;
    const float* __restrict__ Xb = x + b * Nn * Cin + colbase + r;

    v8f acc = {};
    // 76 unconditional steps: K = 0..303
    for (int k0 = 0; k0 < 304; k0 += 4) {
        const int k = k0 + 2 * half;
        v2f a = *(const v2f*)&As[r][k];          // one ds_load_b64
        v2f bb;
        bb.x = Xb[k * Cin];
        bb.y = Xb[(k + 1) * Cin];
        acc = __builtin_amdgcn_wmma_f32_16x16x4_f32(false, a, false, bb,
                                                    (short)0, acc, false, false);
    }
    // peeled tail: K = 304..307 (As col 307 is zero; clamp B addr branchlessly)
    {
        const int k = 304 + 2 * half;
        v2f a = *(const v2f*)&As[r][k];
        const int kc0 = k < Nn ? k : (Nn - 1);
        const int kc1 = (k + 1) < Nn ? (k + 1) : (Nn - 1);
        v2f bb;
        bb.x = Xb[kc0 * Cin];
        bb.y = Xb[kc1 * Cin];
        acc = __builtin_amdgcn_wmma_f32_16x16x4_f32(false, a, false, bb,
                                                    (short)0, acc, false, false);
    }

#pragma unroll
    for (int v = 0; v < 8; ++v) {
        const int row_l = v + 8 * half;
        const int row_g = mtile * 16 + row_l;
        if (row_g < Nn)
            xg[(b * Nn + row_g) * (3 * Cin) + (ks + 1) * Cin + colbase + r] = acc[v];
    }
}

// ---------------------------------------------------------------------------
// K5: main WMMA GEMM  out[row,o] = sum_{d,j} (emb[row,d]*xg[row,j]) * W[d,j,o] + bias
// [9824 x 1920] x [1920 x 64]; grid 614 blocks x 128 threads (4 waves x 16 cols).
// All dims exact multiples of the tile -> zero conditionals in the hot loop.
// ---------------------------------------------------------------------------
__global__ void main_gemm_kernel(const float* __restrict__ xg, const float* __restrict__ emb,
                                 const float* __restrict__ W, const float* __restrict__ bias_pool,
                                 float* __restrict__ out) {
    __shared__ __align__(16) float xg_s[16][XG_STRIDE];  // 8B rows -> ds_load_b64
    __shared__ float emb_s[16][11];
    const int tile = blockIdx.x;
    const int tid = threadIdx.x;

    for (int idx = tid; idx < 16 * 192; idx += 128) {    // 24 iters exact
        int r = idx / 192, c = idx % 192;
        xg_s[r][c] = xg[(tile * 16 + r) * 192 + c];
    }
    for (int idx = tid; idx < 16 * Dd; idx += 128) {
        int r = idx / Dd, d = idx % Dd;
        emb_s[r][d] = emb[(tile * 16 + r) * Dd + d];
    }
    __syncthreads();

    const int lane = tid & 31, wv = tid >> 5;
    const int half = lane >> 4;
    const int r = lane & 15;
    const int colbase = wv * 16;

    v8f acc = {};
    for (int d = 0; d < Dd; ++d) {
        const float e = emb_s[r][d];
        const float* __restrict__ Wd = W + d * 192 * Cout + colbase + r;
        if (d + 1 < Dd) __builtin_prefetch(Wd + 192 * Cout, 0, 1);  // next d-chunk
#pragma unroll 4
        for (int j = 0; j < 192; j += 4) {               // 48 WMMA per d-chunk
            const int jj = j + 2 * half;
            v2f xv = *(const v2f*)&xg_s[r][jj];          // one ds_load_b64
            v2f a;
            a.x = e * xv.x;
            a.y = e * xv.y;
            v2f bb;
            bb.x = Wd[jj * Cout];
            bb.y = Wd[jj * Cout + Cout];
            acc = __builtin_amdgcn_wmma_f32_16x16x4_f32(false, a, false, bb,
                                                        (short)0, acc, false, false);
        }
    }

    const int col = colbase + r;
#pragma unroll
    for (int v = 0; v < 8; ++v) {
        const int row_l = v + 8 * half;
        float bias = 0.f;
#pragma unroll
        for (int d = 0; d < Dd; ++d) bias += emb_s[row_l][d] * bias_pool[d * Cout + col];
        out[(tile * 16 + row_l) * Cout + col] = acc[v] + bias;
    }
}

// ---------------------------------------------------------------------------
extern "C" void kernel_launch(void* const* d_in, const int* in_sizes, int n_in,
                              void* d_out, int out_size, void* d_ws, size_t ws_size,
                              hipStream_t stream) {
    const float* x     = (const float*)d_in[0];
    const float* adj   = (const float*)d_in[1];
    const float* ne    = (const float*)d_in[2];
    const float* wpool = (const float*)d_in[3];   // [10,3,64,64] == [1920,64] row-major
    const float* bpool = (const float*)d_in[4];   // [10,64]
    const float* fc1w  = (const float*)d_in[5];
    const float* fc1b  = (const float*)d_in[6];
    const float* fc2w  = (const float*)d_in[7];
    const float* fc2b  = (const float*)d_in[8];
    float* out = (float*)d_out;

    float* ws  = (float*)d_ws;
    float* att = ws;                               // 128 floats
    float* emb = ws + 128;                         // 98240 floats
    float* T2  = ws + 128 + Bn * Nn * Dd;          // 94249 floats
    float* xg  = ws + 192640;                      // 9824*192 floats (aligned past T2)

    attn_kernel<<<Bn, 64, 0, stream>>>(x, fc1w, fc1b, fc2w, fc2b, att);
    emb_kernel<<<(Bn * Nn * Dd + 255) / 256, 256, 0, stream>>>(att, ne, emb);
    t2_kernel<<<(Nn * Nn + 255) / 256, 256, 0, stream>>>(adj, T2);
    xg0_kernel<<<(ROWS * Cin + 255) / 256, 256, 0, stream>>>(x, xg);
    dim3 g4((Nn + 15) / 16, 2, Bn);                // 20 x 2 x 32
    spmm_wmma_kernel<<<g4, 128, 0, stream>>>(adj, T2, x, xg);
    main_gemm_kernel<<<MT5, 128, 0, stream>>>(xg, emb, wpool, bpool, out);
}